// MoeSpaceTimeModel_35304631173961
// MI455X (gfx1250) — compile-verified
//
#include <hip/hip_runtime.h>
#include <math.h>

typedef __attribute__((ext_vector_type(2))) float v2f;
typedef __attribute__((ext_vector_type(8))) float v8f;

#define NE   16      // experts
#define HID2 32      // 2*HID
#define OUTD 4

// ---------------------------------------------------------------------------
// Precompute Wc[e] = W1[e] (12x32) @ Wf[e] (32x4)  -> (12x4) per expert.
// No activation between the two layers in the reference, so they fuse exactly.
// ---------------------------------------------------------------------------
__global__ void moe_precompute_wc(const float* __restrict__ W1,
                                  const float* __restrict__ Wf,
                                  float* __restrict__ Wc) {
    int i = blockIdx.x * blockDim.x + threadIdx.x;
    if (i >= NE * 12 * OUTD) return;
    int e = i / (12 * OUTD);
    int r = (i % (12 * OUTD)) / OUTD;
    int o = i % OUTD;
    const float* w1 = W1 + (e * 12 + r) * HID2;   // W1[e][r][:]
    const float* wf = Wf + e * HID2 * OUTD + o;   // Wf[e][:][o]
    float acc = 0.f;
#pragma unroll
    for (int k = 0; k < HID2; ++k) acc += w1[k] * wf[k * OUTD];
    Wc[i] = acc;
}

// ---------------------------------------------------------------------------
// Main fused MoE kernel. One thread = one point. 256 threads = 8 wave32.
// Gating matmul done with V_WMMA_F32_16X16X4_F32 (16 points per WMMA tile).
// ---------------------------------------------------------------------------
__launch_bounds__(256)
__global__ void moe_forward(const float* __restrict__ xyzt,
                            const float* __restrict__ Wg,
                            const float* __restrict__ bg,
                            const float* __restrict__ proj,
                            const float* __restrict__ center,
                            const float* __restrict__ tb1,
                            const float* __restrict__ tb2,
                            const float* __restrict__ Wc,
                            const float* __restrict__ bf,
                            float* __restrict__ out) {
    __shared__ __align__(16) float sA[256 * 4];     // xyzt staging (A fragments)
    __shared__ __align__(16) float sL[256 * 16];    // logits tile (C matrix)
    __shared__ __align__(16) float sProj[NE * 24];  // proj[e][3][8]
    __shared__ __align__(16) float sCen[NE * 4];    // center padded to 4
    __shared__ __align__(16) float sWc[NE * 48];    // fused 12x4 per expert
    __shared__ __align__(16) float sBf[NE * 4];
    __shared__ float sTb1[NE];
    __shared__ float sTb2[NE];

    const int tid = threadIdx.x;
    const int pt  = blockIdx.x * 256 + tid;

    // ---- stage expert tables into LDS (tiny: ~5 KB) ----
    for (int i = tid; i < NE * 24; i += 256) sProj[i] = proj[i];
    for (int i = tid; i < NE * 48; i += 256) sWc[i]  = Wc[i];
    if (tid < NE * 4) sBf[tid] = bf[tid];
    if (tid < NE) {
        sTb1[tid] = tb1[tid];
        sTb2[tid] = tb2[tid];
        sCen[tid * 4 + 0] = center[tid * 3 + 0];
        sCen[tid * 4 + 1] = center[tid * 3 + 1];
        sCen[tid * 4 + 2] = center[tid * 3 + 2];
        sCen[tid * 4 + 3] = 0.f;
    }

    // ---- one coalesced 16B load per point; keep in registers + stage to LDS ----
    const float4 p = *(const float4*)(xyzt + (size_t)pt * 4);
    *(float4*)(sA + tid * 4) = p;

    __syncthreads();

    // ---- gating logits via WMMA f32 16x16x4:  C(16x16) = A(16x4) x B(4x16) + bg ----
    const int wave = tid >> 5;
    const int lane = tid & 31;
    const int half = lane >> 4;     // which K pair this lane holds
    const int lm   = lane & 15;     // M (A) / N (B,C) index
    const float* wA = sA + wave * 128;   // this wave's 32 points
    float*       wL = sL + wave * 512;   // this wave's 32x16 logits

    // B fragment: lane holds Wg[2*half][lm] in v0, Wg[2*half+1][lm] in v1
    v2f bfrag;
    bfrag[0] = Wg[(2 * half + 0) * NE + lm];
    bfrag[1] = Wg[(2 * half + 1) * NE + lm];
    const float bgv = bg[lm];

#pragma unroll
    for (int tile = 0; tile < 2; ++tile) {
        // A fragment: lane holds xyzt[tile*16+lm][2*half] / [2*half+1]
        v2f afrag = *(const v2f*)(wA + (tile * 16 + lm) * 4 + 2 * half);
        v8f c;
#pragma unroll
        for (int j = 0; j < 8; ++j) c[j] = bgv;   // bias broadcast per column N
        c = __builtin_amdgcn_wmma_f32_16x16x4_f32(
                /*neg_a=*/false, afrag, /*neg_b=*/false, bfrag,
                /*c_mod=*/(short)0, c, /*reuse_a=*/false, /*reuse_b=*/false);
        // C layout: VGPR j -> row M = j + 8*half, col N = lm
#pragma unroll
        for (int j = 0; j < 8; ++j)
            wL[(tile * 16 + j + 8 * half) * 16 + lm] = c[j];
    }

    __syncthreads();

    // ---- load my 16 logits as 4x float4, then branchless top-2 scan ----
    const float4* myL4 = (const float4*)(sL + tid * 16);
    float lg[NE];
#pragma unroll
    for (int q = 0; q < 4; ++q) {
        float4 v = myL4[q];
        lg[q * 4 + 0] = v.x; lg[q * 4 + 1] = v.y;
        lg[q * 4 + 2] = v.z; lg[q * 4 + 3] = v.w;
    }

    float l0 = lg[0]; int e0 = 0;
    float l1 = -INFINITY; int e1 = 0;
#pragma unroll
    for (int n = 1; n < NE; ++n) {
        const float v   = lg[n];
        const bool gt0  = v > l0;               // new best
        const bool gt1  = v > l1;               // new runner-up
        const float nl1 = gt0 ? l0 : (gt1 ? v : l1);
        const int   ne1 = gt0 ? e0 : (gt1 ? n : e1);
        l0 = gt0 ? v : l0;
        e0 = gt0 ? n : e0;
        l1 = nl1;
        e1 = ne1;
    }
    const float p0 = 1.f / (1.f + expf(l1 - l0));   // softmax over (l0, l1)
    const float p1 = 1.f - p0;

    // ---- fused expert evaluation (2 selected experts) ----
    float ox = 0.f, oy = 0.f, oz = 0.f, ow = 0.f;
#pragma unroll
    for (int s = 0; s < 2; ++s) {
        const int   e = s ? e1 : e0;
        const float w = s ? p1 : p0;

        const float4 cen = *(const float4*)(sCen + e * 4);
        const float cx = p.x - cen.x;
        const float cy = p.y - cen.y;
        const float cz = p.z - cen.z;
        const float rho   = sqrtf(cx * cx + cy * cy + cz * cz);
        const float phi   = atan2f(cy, cx);
        const float theta = acosf(cz / (rho + 1e-6f));

        const float ct  = cosf(p.w + sTb1[e]);
        const float st  = sinf(p.w + sTb2[e]);
        const float sct = ct / (1.f + expf(-ct));   // silu
        const float sst = st / (1.f + expf(-st));

        // emb = [ (rho,phi,theta) @ proj_e (3x8),  ct, st, silu(ct), silu(st) ]
        float emb[12];
        const float* pr = sProj + e * 24;
#pragma unroll
        for (int j = 0; j < 8; ++j)
            emb[j] = rho * pr[j] + phi * pr[8 + j] + theta * pr[16 + j];
        emb[8] = ct; emb[9] = st; emb[10] = sct; emb[11] = sst;

        // out_e = emb(12) @ Wc_e(12x4) + bf_e
        const float4* wc = (const float4*)(sWc + e * 48);
        float4 acc = *(const float4*)(sBf + e * 4);
#pragma unroll
        for (int i = 0; i < 12; ++i) {
            float4 row = wc[i];
            acc.x += emb[i] * row.x;
            acc.y += emb[i] * row.y;
            acc.z += emb[i] * row.z;
            acc.w += emb[i] * row.w;
        }
        ox += w * acc.x; oy += w * acc.y; oz += w * acc.z; ow += w * acc.w;
    }

    *(float4*)(out + (size_t)pt * 4) = make_float4(ox, oy, oz, ow);
}

// ---------------------------------------------------------------------------
extern "C" void kernel_launch(void* const* d_in, const int* in_sizes, int n_in,
                              void* d_out, int out_size, void* d_ws, size_t ws_size,
                              hipStream_t stream) {
    const float* xyzt   = (const float*)d_in[0];
    const float* Wg     = (const float*)d_in[1];
    const float* bg     = (const float*)d_in[2];
    const float* proj   = (const float*)d_in[3];
    const float* center = (const float*)d_in[4];
    const float* tb1    = (const float*)d_in[5];
    const float* tb2    = (const float*)d_in[6];
    const float* W1     = (const float*)d_in[7];
    const float* Wf     = (const float*)d_in[8];
    const float* bf     = (const float*)d_in[9];
    float* out = (float*)d_out;
    float* Wc  = (float*)d_ws;                 // 16*12*4 = 768 floats of scratch

    const int n = in_sizes[0] / 4;             // 524288 points (divisible by 256)

    moe_precompute_wc<<<(NE * 12 * OUTD + 255) / 256, 256, 0, stream>>>(W1, Wf, Wc);
    moe_forward<<<n / 256, 256, 0, stream>>>(xyzt, Wg, bg, proj, center,
                                             tb1, tb2, Wc, bf, out);
}